// Block_10153302687984
// MI455X (gfx1250) — compile-verified
//
#include <hip/hip_runtime.h>

#define NNODES 50000
#define NEDGES 800000
#define DIM    96
#define ND     (NNODES * DIM)     // 4,800,000 floats
#define NTILES (NNODES / 16)      // 3125 (exact)

typedef __attribute__((ext_vector_type(2))) float v2f;
typedef __attribute__((ext_vector_type(8))) float v8f;

// Async copy of one 16B chunk: global (saddr + voff) -> LDS byte offset ldsoff.
// Tracked with ASYNCcnt (CDNA5 GLOBAL_LOAD_ASYNC_TO_LDS_B128, GVS addressing).
__device__ __forceinline__ void async_g2l_b128(unsigned ldsoff, unsigned goff,
                                               const float* saddr) {
    asm volatile("global_load_async_to_lds_b128 %0, %1, %2"
                 :: "v"(ldsoff), "v"(goff), "s"(saddr)
                 : "memory");
}
__device__ __forceinline__ void wait_asynccnt0() {
    asm volatile("s_wait_asynccnt 0x0" ::: "memory");
}

// ---------------------------------------------------------------------------
// Zero-fill workspace accumulation buffers (agg1, agg2, deg are contiguous).
// ---------------------------------------------------------------------------
__global__ void k_zero(float* __restrict__ p, int n) {
    int i = blockIdx.x * blockDim.x + threadIdx.x;
    int stride = gridDim.x * blockDim.x;
    for (; i < n; i += stride) p[i] = 0.0f;
}

// ---------------------------------------------------------------------------
// Edge gather/scatter: one wave32 per edge. Lane l handles features
// l, l+32, l+64 -> 3 coalesced 128B gather chunks + 3 global fp32 atomics.
// Lane 0 optionally accumulates the destination degree.
// ---------------------------------------------------------------------------
__global__ __launch_bounds__(256)
void k_edge_agg(const float* __restrict__ h,
                const int* __restrict__ src,
                const int* __restrict__ dst,
                float* __restrict__ agg,
                float* __restrict__ deg,   // may be null
                int n_edges) {
    int wave = (blockIdx.x * blockDim.x + threadIdx.x) >> 5;
    int lane = threadIdx.x & 31;
    if (wave >= n_edges) return;
    int s = src[wave];
    int d = dst[wave];
    const float* hs = h + (size_t)s * DIM;
    float* ad = agg + (size_t)d * DIM;
#pragma unroll
    for (int c = 0; c < 3; ++c) {
        atomicAdd(&ad[lane + 32 * c], hs[lane + 32 * c]);
    }
    if (deg != nullptr && lane == 0) {
        atomicAdd(&deg[d], 1.0f);
    }
}

// ---------------------------------------------------------------------------
// Fused dual-GEMM conv layer:
//   Out[m, :] = H[m, :] @ Wroot + (AGG[m, :] * scale_m) @ Wnbr + bias
// One wave per 16-row tile; V_WMMA_F32_16X16X4_F32 over K=96 (24 steps),
// 6 N-tiles of 16. Weights staged into LDS with async global->LDS B128
// copies (ASYNCcnt), bias via normal path. 74,112 B dynamic LDS.
//
// A layout (16x4 fp32, ISA 7.12.2): lane row = lane&15, hi = lane>>4;
//   vgpr0 = A[row][kb+2*hi], vgpr1 = A[row][kb+2*hi+1]   (kb = 4*kstep)
// B layout (4x16): vgpr0 = B[kb+2*hi][n], vgpr1 = B[kb+2*hi+1][n], n = lane&15
// C/D layout: vgpr j = element [M = j + 8*hi][N = lane&15]
// ---------------------------------------------------------------------------
__global__ __launch_bounds__(256)
void k_conv_wmma(const float* __restrict__ H,
                 const float* __restrict__ AGG,
                 const float* __restrict__ deg,   // null => no neighbor norm
                 const float* __restrict__ Wroot,
                 const float* __restrict__ Wnbr,
                 const float* __restrict__ bias,
                 float* __restrict__ Out,
                 int n_tiles) {
    extern __shared__ float smem[];
    float* sWr = smem;                 // DIM*DIM floats @ LDS offset 0
    float* sWn = smem + DIM * DIM;     // DIM*DIM floats @ offset WBYTES
    float* sB  = smem + 2 * DIM * DIM; // DIM floats    @ offset 2*WBYTES
    const unsigned WBYTES = DIM * DIM * 4;       // 36,864
    const int WCHUNKS = (DIM * DIM) / 4;         // 2,304 x 16B per matrix

    // Async-stage both weight matrices straight into LDS (no VGPR round-trip).
    for (int c = threadIdx.x; c < WCHUNKS; c += blockDim.x)
        async_g2l_b128((unsigned)(c * 16), (unsigned)(c * 16), Wroot);
    for (int c = threadIdx.x; c < WCHUNKS; c += blockDim.x)
        async_g2l_b128(WBYTES + (unsigned)(c * 16), (unsigned)(c * 16), Wnbr);
    for (int i = threadIdx.x; i < DIM; i += blockDim.x) sB[i] = bias[i];
    wait_asynccnt0();          // each wave drains its own ASYNCcnt
    __syncthreads();           // then all waves rendezvous

    int wave = threadIdx.x >> 5;
    int lane = threadIdx.x & 31;
    int tile = blockIdx.x * (blockDim.x >> 5) + wave;
    if (tile >= n_tiles) return;              // whole-wave exit: EXEC stays all-1s

    int row = lane & 15;
    int hi  = lane >> 4;
    int m   = tile * 16 + row;
    const float* hrow = H   + (size_t)m * DIM;
    const float* arow = AGG + (size_t)m * DIM;
    float scale = 1.0f;
    if (deg != nullptr) scale = 1.0f / fmaxf(deg[m], 1.0f);

    v8f acc[6];
#pragma unroll
    for (int t = 0; t < 6; ++t) {
        float bv = sB[t * 16 + row];
        v8f c;
#pragma unroll
        for (int j = 0; j < 8; ++j) c[j] = bv;
        acc[t] = c;
    }

    for (int k = 0; k < 24; ++k) {
        int kb = k * 4 + 2 * hi;
        v2f ax, aa;
        ax.x = hrow[kb];
        ax.y = hrow[kb + 1];
        aa.x = arow[kb] * scale;
        aa.y = arow[kb + 1] * scale;
#pragma unroll
        for (int t = 0; t < 6; ++t) {
            int n = t * 16 + row;
            v2f br, bn;
            br.x = sWr[kb * DIM + n];
            br.y = sWr[(kb + 1) * DIM + n];
            bn.x = sWn[kb * DIM + n];
            bn.y = sWn[(kb + 1) * DIM + n];
            acc[t] = __builtin_amdgcn_wmma_f32_16x16x4_f32(
                false, ax, false, br, (short)0, acc[t], false, false);
            acc[t] = __builtin_amdgcn_wmma_f32_16x16x4_f32(
                false, aa, false, bn, (short)0, acc[t], false, false);
        }
    }

#pragma unroll
    for (int t = 0; t < 6; ++t) {
        int n = t * 16 + row;
#pragma unroll
        for (int j = 0; j < 8; ++j) {
            Out[(size_t)(tile * 16 + j + 8 * hi) * DIM + n] = acc[t][j];
        }
    }
}

// ---------------------------------------------------------------------------
// Fused concat + linear + ReLU:
//   out = relu([h1 | h2] @ lin_w^T + lin_b),  lin_w is (DIM, 2*DIM) row-major
// K = 192 walked as 48 WMMA K-steps; k<96 reads h1, else h2.
// B[k][n] = lin_w[n*192 + k] read transposed out of LDS.
// ---------------------------------------------------------------------------
__global__ __launch_bounds__(256)
void k_final_wmma(const float* __restrict__ h1,
                  const float* __restrict__ h2,
                  const float* __restrict__ linw,
                  const float* __restrict__ linb,
                  float* __restrict__ Out,
                  int n_tiles) {
    extern __shared__ float smem[];
    float* sW = smem;                 // DIM * 2*DIM floats @ LDS offset 0
    float* sB = smem + DIM * 2 * DIM; // DIM floats
    const int LCHUNKS = (DIM * 2 * DIM) / 4;     // 4,608 x 16B

    for (int c = threadIdx.x; c < LCHUNKS; c += blockDim.x)
        async_g2l_b128((unsigned)(c * 16), (unsigned)(c * 16), linw);
    for (int i = threadIdx.x; i < DIM; i += blockDim.x) sB[i] = linb[i];
    wait_asynccnt0();
    __syncthreads();

    int wave = threadIdx.x >> 5;
    int lane = threadIdx.x & 31;
    int tile = blockIdx.x * (blockDim.x >> 5) + wave;
    if (tile >= n_tiles) return;

    int row = lane & 15;
    int hi  = lane >> 4;
    int m   = tile * 16 + row;
    const float* r1 = h1 + (size_t)m * DIM;
    const float* r2 = h2 + (size_t)m * DIM;

    v8f acc[6];
#pragma unroll
    for (int t = 0; t < 6; ++t) {
        float bv = sB[t * 16 + row];
        v8f c;
#pragma unroll
        for (int j = 0; j < 8; ++j) c[j] = bv;
        acc[t] = c;
    }

    for (int k = 0; k < 48; ++k) {
        int kb = k * 4 + 2 * hi;
        const float* srcrow = (kb < DIM) ? (r1 + kb) : (r2 + (kb - DIM));
        v2f ax;
        ax.x = srcrow[0];
        ax.y = srcrow[1];
#pragma unroll
        for (int t = 0; t < 6; ++t) {
            int n = t * 16 + row;
            v2f b;
            b.x = sW[n * (2 * DIM) + kb];
            b.y = sW[n * (2 * DIM) + kb + 1];
            acc[t] = __builtin_amdgcn_wmma_f32_16x16x4_f32(
                false, ax, false, b, (short)0, acc[t], false, false);
        }
    }

#pragma unroll
    for (int t = 0; t < 6; ++t) {
        int n = t * 16 + row;
#pragma unroll
        for (int j = 0; j < 8; ++j) {
            Out[(size_t)(tile * 16 + j + 8 * hi) * DIM + n] = fmaxf(acc[t][j], 0.0f);
        }
    }
}

// ---------------------------------------------------------------------------
extern "C" void kernel_launch(void* const* d_in, const int* in_sizes, int n_in,
                              void* d_out, int out_size, void* d_ws, size_t ws_size,
                              hipStream_t stream) {
    const float* x   = (const float*)d_in[0];
    const int*   src = (const int*)d_in[1];
    const int*   dst = src + NEDGES;
    const float* W1r = (const float*)d_in[2];
    const float* W1n = (const float*)d_in[3];
    const float* b1  = (const float*)d_in[4];
    const float* W2r = (const float*)d_in[5];
    const float* W2n = (const float*)d_in[6];
    const float* b2  = (const float*)d_in[7];
    const float* lw  = (const float*)d_in[8];
    const float* lb  = (const float*)d_in[9];
    float* out = (float*)d_out;

    // workspace layout: [agg1 | agg2 | deg | h1 | h2]
    float* agg1 = (float*)d_ws;
    float* agg2 = agg1 + ND;
    float* deg  = agg2 + ND;
    float* h1   = deg + NNODES;
    float* h2   = h1 + ND;

    // 1) zero accumulation buffers (agg1, agg2, deg are contiguous)
    k_zero<<<2048, 256, 0, stream>>>(agg1, 2 * ND + NNODES);

    int edge_blocks = (NEDGES + 7) / 8;        // 8 waves (edges) per 256-thr block
    int conv_blocks = (NTILES + 7) / 8;        // 8 row-tiles per block
    size_t conv_lds = (size_t)(2 * DIM * DIM + DIM) * sizeof(float);  // 74,112 B
    size_t fin_lds  = (size_t)(DIM * 2 * DIM + DIM) * sizeof(float);  // 74,112 B

    // 2) layer-1 aggregation (+degree, reused by layer 2)
    k_edge_agg<<<edge_blocks, 256, 0, stream>>>(x, src, dst, agg1, deg, NEDGES);

    // 3) h1 = x @ W1_root + agg1 @ W1_nbr + b1
    k_conv_wmma<<<conv_blocks, 256, conv_lds, stream>>>(
        x, agg1, nullptr, W1r, W1n, b1, h1, NTILES);

    // 4) layer-2 aggregation
    k_edge_agg<<<edge_blocks, 256, 0, stream>>>(h1, src, dst, agg2, nullptr, NEDGES);

    // 5) h2 = h1 @ W2_root + (agg2/deg) @ W2_nbr + b2
    k_conv_wmma<<<conv_blocks, 256, conv_lds, stream>>>(
        h1, agg2, deg, W2r, W2n, b2, h2, NTILES);

    // 6) out = relu([h1|h2] @ lin_w^T + lin_b)
    k_final_wmma<<<conv_blocks, 256, fin_lds, stream>>>(h1, h2, lw, lb, out, NTILES);
}